// DescriptorBuilder_11338713661482
// MI455X (gfx1250) — compile-verified
//
#include <hip/hip_runtime.h>

// CDNA5 / gfx1250: wave32, WMMA 16x16x4 f32 for the batched dr·drT GEMM.

typedef __attribute__((ext_vector_type(2))) float v2f;
typedef __attribute__((ext_vector_type(8))) float v8f;

#define NATOMS   192
#define NTILE    12            // 192 / 16
#define RC_F     5.0f
#define NFEAT_R  9             // rij^0..rij^8
#define NFEAT_A  9             // n in 0..2  x  l in 0..2
#define NFEAT    (NFEAT_R + NFEAT_A)
#define PI_F     3.14159265358979f

// fast reciprocal: v_rcp_f32 + one Newton step (~1 ulp), avoids the
// v_div_scale/v_div_fmas IEEE division chain the compiler emits for '/'
__device__ __forceinline__ float fast_rcp(float x) {
    float r = __builtin_amdgcn_rcpf(x);
    r = r * (2.0f - x * r);
    return r;
}

__global__ __launch_bounds__(256) void desc_kernel(const float* __restrict__ R,
                                                   const float* __restrict__ box,
                                                   float* __restrict__ out)
{
    __shared__ float s_drx[NATOMS], s_dry[NATOMS], s_drz[NATOMS];
    __shared__ float s_r[NATOMS],   s_w[NATOMS];      // s_w = fc * mask
    __shared__ float s_acc[NFEAT];

    const int i    = blockIdx.x;
    const int tid  = threadIdx.x;
    const int lane = tid & 31;     // wave32
    const int wave = tid >> 5;     // 8 waves / block

    if (tid < NFEAT) s_acc[tid] = 0.0f;

    // ---- box inverse (general 3x3, computed per-thread in registers) ----
    float b00 = box[0], b01 = box[1], b02 = box[2];
    float b10 = box[3], b11 = box[4], b12 = box[5];
    float b20 = box[6], b21 = box[7], b22 = box[8];
    float det = b00 * (b11 * b22 - b12 * b21)
              - b01 * (b10 * b22 - b12 * b20)
              + b02 * (b10 * b21 - b11 * b20);
    float id  = fast_rcp(det);
    float i00 =  (b11 * b22 - b12 * b21) * id;
    float i01 = -(b01 * b22 - b02 * b21) * id;
    float i02 =  (b01 * b12 - b02 * b11) * id;
    float i10 = -(b10 * b22 - b12 * b20) * id;
    float i11 =  (b00 * b22 - b02 * b20) * id;
    float i12 = -(b00 * b12 - b02 * b10) * id;
    float i20 =  (b10 * b21 - b11 * b20) * id;
    float i21 = -(b00 * b21 - b01 * b20) * id;
    float i22 =  (b00 * b11 - b01 * b10) * id;

    float Rix = R[3 * i + 0], Riy = R[3 * i + 1], Riz = R[3 * i + 2];
    float six = i00 * Rix + i01 * Riy + i02 * Riz;
    float siy = i10 * Rix + i11 * Riy + i12 * Riz;
    float siz = i20 * Rix + i21 * Riy + i22 * Riz;

    // ---- pass A: per-pair (i, j=tid) geometry into LDS + radial features ----
    float qr[NFEAT_R];
    #pragma unroll
    for (int k = 0; k < NFEAT_R; ++k) qr[k] = 0.0f;

    if (tid < NATOMS) {
        int   j   = tid;
        float Rjx = R[3 * j + 0], Rjy = R[3 * j + 1], Rjz = R[3 * j + 2];
        float sjx = i00 * Rjx + i01 * Rjy + i02 * Rjz;
        float sjy = i10 * Rjx + i11 * Rjy + i12 * Rjz;
        float sjz = i20 * Rjx + i21 * Rjy + i22 * Rjz;
        float dsx = six - sjx; dsx -= rintf(dsx);   // round-half-even = jnp.round
        float dsy = siy - sjy; dsy -= rintf(dsy);
        float dsz = siz - sjz; dsz -= rintf(dsz);
        float dx  = b00 * dsx + b01 * dsy + b02 * dsz;
        float dy  = b10 * dsx + b11 * dsy + b12 * dsz;
        float dz  = b20 * dsx + b21 * dsy + b22 * dsz;
        float r   = sqrtf(dx * dx + dy * dy + dz * dz);
        float x   = fminf(r * (1.0f / RC_F), 1.0f);
        float fc  = 0.5f * (cosf(PI_F * x) + 1.0f);
        float w   = (j == i) ? 0.0f : fc;           // fc * mask
        s_drx[j] = dx; s_dry[j] = dy; s_drz[j] = dz;
        s_r[j] = r;    s_w[j] = w;
        float p = w;                                 // w * r^0
        #pragma unroll
        for (int k = 0; k < NFEAT_R; ++k) { qr[k] = p; p *= r; }
    }

    // radial reduction: waves 0..5 fully cover j = 0..191
    if (wave < 6) {
        #pragma unroll
        for (int k = 0; k < NFEAT_R; ++k) {
            float v = qr[k];
            #pragma unroll
            for (int s = 16; s >= 1; s >>= 1) v += __shfl_xor(v, s, 32);
            if (lane == 0) atomicAdd(&s_acc[k], v);
        }
    }
    __syncthreads();

    // ---- pass B: angular features via WMMA f32 16x16x4 over (j,k) tiles ----
    float qa[NFEAT_A];
    #pragma unroll
    for (int m = 0; m < NFEAT_A; ++m) qa[m] = 0.0f;

    const int hi  = lane >> 4;   // 0: K=0,1 half | 1: K=2,3 half
    const int l15 = lane & 15;

    for (int t = wave; t < NTILE * NTILE; t += 8) {   // 18 tiles/wave, uniform
        int ja = (t / NTILE) * 16 + l15;              // A row  (j index)
        int ka = (t % NTILE) * 16 + l15;              // B col  (k index)

        v2f a, b;
        a.x = hi ? s_drz[ja] : s_drx[ja];             // A: K=0/2
        a.y = hi ? 0.0f      : s_dry[ja];             // A: K=1/3 (pad)
        b.x = hi ? s_drz[ka] : s_drx[ka];             // B: K=0/2
        b.y = hi ? 0.0f      : s_dry[ka];             // B: K=1/3 (pad)

        v8f c = {};
        c = __builtin_amdgcn_wmma_f32_16x16x4_f32(
                false, a, false, b, (short)0, c, false, false);
        // c[v] = dot(dr_j, dr_k) at  j = jb*16 + v + 8*hi,  k = kb*16 + l15

        float rk = s_r[ka];
        float wk = s_w[ka];
        int   jr = (t / NTILE) * 16 + (hi ? 8 : 0);
        #pragma unroll
        for (int v = 0; v < 8; ++v) {
            float rj    = s_r[jr + v];
            float wj    = s_w[jr + v];
            float dot   = c[v];
            float denom = rj * rk + 1e-8f;
            float ct    = dot * fast_rcp(denom);
            ct          = fminf(fmaxf(ct, -1.0f), 1.0f);
            float p2    = 0.5f * (3.0f * ct * ct - 1.0f);
            float base  = wj * wk;
            float rr    = rj * rk;
            float rn    = base;                        // base * rr^n
            #pragma unroll
            for (int n = 0; n < 3; ++n) {
                qa[n * 3 + 0] += rn;                   // P0 = 1
                qa[n * 3 + 1] += rn * ct;              // P1 = cos
                qa[n * 3 + 2] += rn * p2;              // P2
                rn *= rr;
            }
        }
    }

    // angular reduction: shuffle tree within wave, LDS float atomics across waves
    #pragma unroll
    for (int m = 0; m < NFEAT_A; ++m) {
        float v = qa[m];
        #pragma unroll
        for (int s = 16; s >= 1; s >>= 1) v += __shfl_xor(v, s, 32);
        if (lane == 0) atomicAdd(&s_acc[NFEAT_R + m], v);
    }
    __syncthreads();

    if (tid < NFEAT) out[i * NFEAT + tid] = s_acc[tid];
}

extern "C" void kernel_launch(void* const* d_in, const int* in_sizes, int n_in,
                              void* d_out, int out_size, void* d_ws, size_t ws_size,
                              hipStream_t stream) {
    const float* R   = (const float*)d_in[0];   // (192, 3) f32
    const float* box = (const float*)d_in[1];   // (3, 3)  f32
    float*       out = (float*)d_out;           // (192, 18) f32
    desc_kernel<<<NATOMS, 256, 0, stream>>>(R, box, out);
}